// SparseBlockSeries_8581344657793
// MI455X (gfx1250) — compile-verified
//
#include <hip/hip_runtime.h>

typedef __attribute__((ext_vector_type(2))) float v2f;
typedef __attribute__((ext_vector_type(8))) float v8f;

#define N_SITES 131072
#define CCH     128
#define KTAP    9
#define NBLK    4
#define EPSV    1e-5f

#define MT      64      // rows per workgroup
#define LXS     132     // padded LDS row stride (floats): bank = 4*n16 + col -> conflict-free A reads
#define NWG     (N_SITES / MT)   // 2048 workgroups per conv

// ---------------------------------------------------------------------------
// Submanifold sparse conv: out[n,:] = sum_k gather(x, idx[n,k]) @ W[k]
// One WG: 64 rows x 128 cols, 8 waves. Wave w owns col tile w (16 cols) and all
// 4 row strips -> 4x v8f acc; B frag loaded once per kk, reused by 4 WMMAs.
// Gather is double-buffered via GLOBAL_LOAD_ASYNC_TO_LDS_B128 (ASYNCcnt);
// invalid neighbors stream from a zeroed row (branch-free).
// Fused deterministic BN-stat partials (sum, sumsq per channel) -> gpart.
// ---------------------------------------------------------------------------
__global__ __launch_bounds__(256, 2)
void subm_conv_wmma(const float* __restrict__ xin,
                    const int*   __restrict__ nidx,
                    const float* __restrict__ W,      // [KTAP, C, C]
                    const float* __restrict__ zrow,   // [C] zeros
                    float*       __restrict__ out,    // [N, C]
                    float*       __restrict__ gpart)  // [NWG, 256] per-WG (sum[128], sq[128])
{
    __shared__ float lx[2][MT * LXS];       // double-buffered gathered activations
    __shared__ float lps[16 * 16];          // per-(wave,khalf) channel partial sums
    __shared__ float lpq[16 * 16];          // per-(wave,khalf) channel partial sumsq

    const int tid   = threadIdx.x;
    const int lane  = tid & 31;
    const int wave  = tid >> 5;             // col tile 0..7
    const int n16   = lane & 15;
    const int khalf = lane >> 4;
    const int m0    = blockIdx.x * MT;

    v8f acc[4] = {};

    // gather mapping: 4 threads per row, 32 floats (8 x float4) each
    const int gr = tid >> 2;
    const int gq = tid & 3;
    const unsigned ldsb0 = (unsigned)(unsigned long long)&lx[0][gr * LXS + gq * 32];
    const unsigned ldsb1 = (unsigned)(unsigned long long)&lx[1][gr * LXS + gq * 32];

    // ---- issue async gather for tap k into buffer (k&1) ----
    auto issue_gather = [&](int k, unsigned ldsb) {
        const int src = nidx[(m0 + gr) * KTAP + k];
        const float* xr = (src >= 0) ? (xin + (size_t)src * CCH + gq * 32)
                                     : (zrow + gq * 32);
        #pragma unroll
        for (int j = 0; j < 8; ++j) {
            unsigned long long ga = (unsigned long long)(const void*)(xr + j * 4);
            unsigned la = ldsb + j * 16;
            asm volatile("global_load_async_to_lds_b128 %0, %1, off"
                         :: "v"(la), "v"(ga) : "memory");
        }
    };

    issue_gather(0, ldsb0);

    for (int k = 0; k < KTAP; ++k) {
        // wait for this tap's async DMA, then make it visible to all waves
        asm volatile("s_wait_asynccnt 0x0" ::: "memory");
        __syncthreads();

        // prefetch next tap: async gather into the other buffer (its previous
        // readers finished before the barrier above), + weight panel prefetch
        if (k + 1 < KTAP) {
            issue_gather(k + 1, (k & 1) ? ldsb0 : ldsb1);
            __builtin_prefetch(W + (size_t)(k + 1) * CCH * CCH + tid * 64, 0, 0);
        }

        const float* lbuf = lx[k & 1];
        const float* Wk   = W + (size_t)k * CCH * CCH;
        const float* bp   = Wk + (size_t)(khalf * 2) * CCH + wave * 16 + n16;

        #pragma unroll 4
        for (int kk = 0; kk < 32; ++kk) {
            v2f b;                          // B 4x16 frag: lane(N=n16), K = 4kk + 2*khalf + {0,1}
            b.x = bp[(size_t)kk * 4 * CCH];
            b.y = bp[(size_t)kk * 4 * CCH + CCH];
            #pragma unroll
            for (int st = 0; st < 4; ++st) {
                v2f a = *(const v2f*)&lbuf[(st * 16 + n16) * LXS + kk * 4 + khalf * 2];
                acc[st] = __builtin_amdgcn_wmma_f32_16x16x4_f32(
                    false, a, false, b, (short)0, acc[st], false, false);
            }
        }
        __syncthreads();                    // all waves done with buffer (k&1)
    }

    // ---- epilogue: store D tiles + deterministic channel partials ----
    const int c = wave * 16 + n16;          // this lane's output channel
    float s = 0.f, q2 = 0.f;
    #pragma unroll
    for (int st = 0; st < 4; ++st) {
        const int rowb = m0 + st * 16 + khalf * 8;   // C/D: VGPR v holds M = v + 8*khalf
        #pragma unroll
        for (int v = 0; v < 8; ++v) {
            float val = acc[st][v];
            out[(size_t)(rowb + v) * CCH + c] = val;
            s  += val;
            q2 += val * val;
        }
    }
    lps[(wave * 2 + khalf) * 16 + n16] = s;
    lpq[(wave * 2 + khalf) * 16 + n16] = q2;
    __syncthreads();
    if (tid < CCH) {
        const int w = tid >> 4, cc = tid & 15;
        const float su = lps[(w * 2) * 16 + cc] + lps[(w * 2 + 1) * 16 + cc];
        const float qu = lpq[(w * 2) * 16 + cc] + lpq[(w * 2 + 1) * 16 + cc];
        gpart[(size_t)blockIdx.x * 256 + tid]       = su;
        gpart[(size_t)blockIdx.x * 256 + CCH + tid] = qu;
    }
}

// Reduce per-WG partials -> gstats[256] = {sum[128], sumsq[128]}. Fixed tree: deterministic.
__global__ __launch_bounds__(256)
void reduce_stats(const float* __restrict__ gpart, float* __restrict__ gstats)
{
    __shared__ float sh[256];
    const int b = blockIdx.x;               // stat slot 0..255
    const int t = threadIdx.x;
    float acc = 0.f;
    #pragma unroll
    for (int j = 0; j < NWG / 256; ++j)
        acc += gpart[(size_t)(j * 256 + t) * 256 + b];
    sh[t] = acc;
    __syncthreads();
    for (int off = 128; off > 0; off >>= 1) {
        if (t < off) sh[t] += sh[t + off];
        __syncthreads();
    }
    if (t == 0) gstats[b] = sh[0];
}

// gstats -> per-channel affine: scale = gamma*rsqrt(var+eps), shift = beta - mean*scale
__global__ void bn_finalize(const float* __restrict__ gstats,
                            const float* __restrict__ gamma,
                            const float* __restrict__ beta,
                            float* __restrict__ ss)
{
    const int c = threadIdx.x;              // 128 threads
    const float inv_n = 1.0f / (float)N_SITES;
    const float mean = gstats[c] * inv_n;
    const float var  = gstats[CCH + c] * inv_n - mean * mean;
    const float rs   = rsqrtf(var + EPSV);
    const float sc   = gamma[c] * rs;
    ss[c]       = sc;
    ss[CCH + c] = beta[c] - mean * sc;
}

// y = relu(h*scale + shift [+ res]); float4 per thread; in-place safe (elementwise).
__global__ __launch_bounds__(256)
void bn_apply(const float* __restrict__ h, const float* __restrict__ ss,
              const float* __restrict__ res, float* __restrict__ outp)
{
    const size_t base = ((size_t)blockIdx.x * 256 + threadIdx.x) * 4;
    const int c0 = (int)(base & (size_t)(CCH - 1));
    float4 v = *(const float4*)(h + base);
    float o0 = v.x * ss[c0 + 0] + ss[CCH + c0 + 0];
    float o1 = v.y * ss[c0 + 1] + ss[CCH + c0 + 1];
    float o2 = v.z * ss[c0 + 2] + ss[CCH + c0 + 2];
    float o3 = v.w * ss[c0 + 3] + ss[CCH + c0 + 3];
    if (res) {
        float4 r = *(const float4*)(res + base);
        o0 += r.x; o1 += r.y; o2 += r.z; o3 += r.w;
    }
    o0 = fmaxf(o0, 0.f); o1 = fmaxf(o1, 0.f);
    o2 = fmaxf(o2, 0.f); o3 = fmaxf(o3, 0.f);
    *(float4*)(outp + base) = make_float4(o0, o1, o2, o3);
}

// zero helper (zero row for invalid gathers)
__global__ void zero_buf(float* __restrict__ p) { p[threadIdx.x] = 0.f; }

// ---------------------------------------------------------------------------
extern "C" void kernel_launch(void* const* d_in, const int* in_sizes, int n_in,
                              void* d_out, int out_size, void* d_ws, size_t ws_size,
                              hipStream_t stream)
{
    (void)in_sizes; (void)n_in; (void)out_size; (void)ws_size;

    const float* x      = (const float*)d_in[0];
    const float* W1     = (const float*)d_in[1];   // [B,K,C,C]
    const float* W2     = (const float*)d_in[2];
    const float* gamma1 = (const float*)d_in[3];   // [B,C]
    const float* beta1  = (const float*)d_in[4];
    const float* gamma2 = (const float*)d_in[5];
    const float* beta2  = (const float*)d_in[6];
    const int*   nidx   = (const int*)d_in[7];     // [N,K]
    float*       xout   = (float*)d_out;           // [N,C], also x ping buffer

    const size_t NC = (size_t)N_SITES * CCH;
    float* ws     = (float*)d_ws;
    float* h      = ws;                            // [N,C]
    float* h2     = ws + NC;                       // [N,C]
    float* gpart  = ws + 2 * NC;                   // [NWG,256]
    float* gstats = gpart + (size_t)NWG * 256;     // [256]
    float* ssb    = gstats + 256;                  // [256]
    float* zrow   = ssb + 256;                     // [128] zeros

    const int elem_blocks = (int)(NC / (256 * 4)); // 16384

    zero_buf<<<1, 128, 0, stream>>>(zrow);

    for (int i = 0; i < NBLK; ++i) {
        const float* xin = (i == 0) ? x : (const float*)xout;
        const size_t wo  = (size_t)i * KTAP * CCH * CCH;

        // conv1 + stats -> BN+ReLU in place on h
        subm_conv_wmma<<<NWG, 256, 0, stream>>>(xin, nidx, W1 + wo, zrow, h, gpart);
        reduce_stats<<<256, 256, 0, stream>>>(gpart, gstats);
        bn_finalize<<<1, 128, 0, stream>>>(gstats, gamma1 + i * CCH, beta1 + i * CCH, ssb);
        bn_apply<<<elem_blocks, 256, 0, stream>>>(h, ssb, nullptr, h);

        // conv2 + stats -> BN + residual + ReLU -> xout
        subm_conv_wmma<<<NWG, 256, 0, stream>>>(h, nidx, W2 + wo, zrow, h2, gpart);
        reduce_stats<<<256, 256, 0, stream>>>(gpart, gstats);
        bn_finalize<<<1, 128, 0, stream>>>(gstats, gamma2 + i * CCH, beta2 + i * CCH, ssb);
        bn_apply<<<elem_blocks, 256, 0, stream>>>(h2, ssb, xin, xout);
    }
}